// Attention_6940667150357
// MI455X (gfx1250) — compile-verified
//
#include <hip/hip_runtime.h>
#include <math.h>

// Problem constants (reference: B=64, T_K=1024, H=512, N=2H=1024)
#define BB 64
#define TK 1024
#define NN 1024

typedef __attribute__((ext_vector_type(2))) float v2f;
typedef __attribute__((ext_vector_type(8))) float v8f;

// ---------------------------------------------------------------------------
// Kernel 1: dec_fea[m,n] = sum_k s_t_hat[m,k] * W_d[n,k] + b_d[n]
// (64 x 1024) = (64 x 1024) @ (1024 x 1024)^T, via V_WMMA_F32_16X16X4_F32.
// One wave per 16x16 output tile; 256 tiles total; 4 waves per block.
//
// fp32 WMMA fragment layout (CDNA5 ISA 7.12.2):
//   A (16x4):  lanes 0-15 hold M=lane, VGPR0=K0,VGPR1=K1; lanes 16-31 K2,K3
//   B (4x16):  mirrored (lane half selects K pair), N = lane&15
//   C/D:       VGPR r -> M = r + 8*(lane>=16), N = lane&15
// ---------------------------------------------------------------------------
__global__ __launch_bounds__(128) void k_decproj(const float* __restrict__ S,
                                                 const float* __restrict__ Wd,
                                                 const float* __restrict__ bd,
                                                 float* __restrict__ dec) {
    const int wave = threadIdx.x >> 5;
    const int lane = threadIdx.x & 31;
    const int tile = blockIdx.x * 4 + wave;       // 0..255
    const int tm   = tile & 3;                    // B/16 = 4 row tiles
    const int tn   = tile >> 2;                   // N/16 = 64 col tiles
    const int half = lane >> 4;                   // K-pair selector
    const int lm   = lane & 15;
    const int m0   = tm * 16;
    const int n0   = tn * 16;
    const int n    = n0 + lm;

    const float* __restrict__ arow = S  + (m0 + lm) * NN;  // A row (M = lm)
    const float* __restrict__ brow = Wd + n * NN;          // B col n == W_d row n

    v8f acc = {};
    for (int k0 = 0; k0 < NN; k0 += 4) {
        const int kk = k0 + 2 * half;
        v2f a = *(const v2f*)(arow + kk);   // A[lm, kk..kk+1]
        v2f b = *(const v2f*)(brow + kk);   // B[kk..kk+1, n] = W_d[n, kk..kk+1]
        acc = __builtin_amdgcn_wmma_f32_16x16x4_f32(
            /*neg_a=*/false, a, /*neg_b=*/false, b,
            /*c_mod=*/(short)0, acc, /*reuse_a=*/false, /*reuse_b=*/false);
    }

    const float bias = bd[n];
#pragma unroll
    for (int r = 0; r < 8; ++r) {
        const int m = m0 + r + 8 * half;
        dec[m * NN + n] = acc[r] + bias;
    }
}

// ---------------------------------------------------------------------------
// Kernel 2: scores[b,t] = sum_n tanh(ef[b,t,n] + dec[b,n] + cov[b,t]*Wc[n])*v[n]
// One 256-thread block per (b,t) row; each thread does one float4 chunk.
// Streams 256 MB of encoder_feature -> bandwidth-bound pass #1.
// ---------------------------------------------------------------------------
__global__ __launch_bounds__(256) void k_scores(const float* __restrict__ ef,
                                                const float* __restrict__ dec,
                                                const float* __restrict__ cov,
                                                const float* __restrict__ Wc,
                                                const float* __restrict__ vvec,
                                                float* __restrict__ scores) {
    const int bt  = blockIdx.x;        // 0 .. B*TK-1
    const int b   = bt >> 10;          // bt / TK
    const int tid = threadIdx.x;
    const int c4  = tid * 4;

    const float  c  = cov[bt];
    const float4 e  = *(const float4*)(ef  + (size_t)bt * NN + c4);
    const float4 d  = *(const float4*)(dec + b * NN + c4);
    const float4 w  = *(const float4*)(Wc  + c4);
    const float4 vv = *(const float4*)(vvec + c4);

    float s = tanhf(e.x + d.x + c * w.x) * vv.x
            + tanhf(e.y + d.y + c * w.y) * vv.y
            + tanhf(e.z + d.z + c * w.z) * vv.z
            + tanhf(e.w + d.w + c * w.w) * vv.w;

    // wave32 reduce
#pragma unroll
    for (int off = 16; off > 0; off >>= 1) s += __shfl_xor(s, off, 32);

    __shared__ float red[8];
    if ((tid & 31) == 0) red[tid >> 5] = s;
    __syncthreads();
    if (tid == 0) {
        float t = 0.f;
#pragma unroll
        for (int i = 0; i < 8; ++i) t += red[i];
        scores[bt] = t;
    }
}

// ---------------------------------------------------------------------------
// Kernel 3: masked softmax + renorm + coverage update, one block per b.
// softmax->mask->renorm folds exactly to: attn = e*mask / sum(e*mask).
// ---------------------------------------------------------------------------
__global__ __launch_bounds__(1024) void k_softmax(const float* __restrict__ scores,
                                                  const float* __restrict__ mask,
                                                  const float* __restrict__ cov,
                                                  float* __restrict__ attn,
                                                  float* __restrict__ covnext) {
    const int b = blockIdx.x;
    const int t = threadIdx.x;
    __shared__ float red[32];
    __shared__ float bcast;

    const float s = scores[b * TK + t];

    // --- block max ---
    float m = s;
#pragma unroll
    for (int off = 16; off > 0; off >>= 1) m = fmaxf(m, __shfl_xor(m, off, 32));
    if ((t & 31) == 0) red[t >> 5] = m;
    __syncthreads();
    if (t < 32) {
        float mm = red[t];
#pragma unroll
        for (int off = 16; off > 0; off >>= 1) mm = fmaxf(mm, __shfl_xor(mm, off, 32));
        if (t == 0) bcast = mm;
    }
    __syncthreads();
    const float gmax = bcast;

    // --- masked exp + block sum ---
    const float w = expf(s - gmax) * mask[b * TK + t];
    float ss = w;
#pragma unroll
    for (int off = 16; off > 0; off >>= 1) ss += __shfl_xor(ss, off, 32);
    if ((t & 31) == 0) red[t >> 5] = ss;
    __syncthreads();
    if (t < 32) {
        float s2 = red[t];
#pragma unroll
        for (int off = 16; off > 0; off >>= 1) s2 += __shfl_xor(s2, off, 32);
        if (t == 0) bcast = s2;
    }
    __syncthreads();

    const float a = w / bcast;
    attn[b * TK + t]    = a;
    covnext[b * TK + t] = cov[b * TK + t] + a;
}

// ---------------------------------------------------------------------------
// Kernel 4: c_t[b,n] = sum_t attn[b,t] * enc_out[b,t,n]
// Each thread owns one n (coalesced across the block per t-step); attn read as
// uniform float4 broadcasts. Streams 256 MB -> bandwidth-bound pass #2.
// ---------------------------------------------------------------------------
__global__ __launch_bounds__(256) void k_ctx(const float* __restrict__ enc,
                                             const float* __restrict__ attn,
                                             float* __restrict__ ct) {
    const int b = blockIdx.y;
    const int n = blockIdx.x * 256 + threadIdx.x;
    const float* __restrict__ ebase = enc + (size_t)b * TK * NN + n;
    const float* __restrict__ arow  = attn + b * TK;

    float acc = 0.f;
    for (int t = 0; t < TK; t += 4) {
        const float4 a = *(const float4*)(arow + t);
        const int tp = (t + 8 < TK) ? (t + 8) : (TK - 1);
        __builtin_prefetch(ebase + (size_t)tp * NN, 0, 1);  // global_prefetch_b8
        acc += a.x * ebase[(size_t)(t + 0) * NN]
             + a.y * ebase[(size_t)(t + 1) * NN]
             + a.z * ebase[(size_t)(t + 2) * NN]
             + a.w * ebase[(size_t)(t + 3) * NN];
    }
    ct[b * NN + n] = acc;
}

// ---------------------------------------------------------------------------
extern "C" void kernel_launch(void* const* d_in, const int* in_sizes, int n_in,
                              void* d_out, int out_size, void* d_ws, size_t ws_size,
                              hipStream_t stream) {
    const float* s_t_hat  = (const float*)d_in[0];   // (B, N)
    const float* enc_out  = (const float*)d_in[1];   // (B, TK, N)
    const float* enc_feat = (const float*)d_in[2];   // (B*TK, N)
    const float* mask     = (const float*)d_in[3];   // (B, TK)
    const float* cov      = (const float*)d_in[4];   // (B, TK)
    const float* Wd       = (const float*)d_in[5];   // (N, N)
    const float* bd       = (const float*)d_in[6];   // (N,)
    const float* Wc       = (const float*)d_in[7];   // (N,)
    const float* v        = (const float*)d_in[8];   // (N,)

    float* out     = (float*)d_out;
    float* ct      = out;                 // (B, N)
    float* attn    = out + BB * NN;       // (B, TK)  [TK == NN == 1024]
    float* covnext = out + 2 * BB * NN;   // (B, TK)

    float* dec    = (float*)d_ws;         // (B, N)   scratch
    float* scores = dec + BB * NN;        // (B, TK)  scratch

    // 1. decoder projection GEMM (WMMA f32)
    k_decproj<<<64, 128, 0, stream>>>(s_t_hat, Wd, bd, dec);
    // 2. additive-attention scores (streams encoder_feature)
    k_scores<<<BB * TK, 256, 0, stream>>>(enc_feat, dec, cov, Wc, v, scores);
    // 3. masked softmax + renorm + coverage update
    k_softmax<<<BB, 1024, 0, stream>>>(scores, mask, cov, attn, covnext);
    // 4. context vector (streams encoder_outputs)
    dim3 g4(NN / 256, BB);
    k_ctx<<<g4, 256, 0, stream>>>(enc_out, attn, ct);
}